// Net_43018392437221
// MI455X (gfx1250) — compile-verified
//
#include <hip/hip_runtime.h>

typedef float v2f __attribute__((ext_vector_type(2)));
typedef float v8f __attribute__((ext_vector_type(8)));

__device__ __forceinline__ void ry_pair(float& a, float& b, float c, float sn) {
    float na = c * a - sn * b;
    float nb = sn * a + c * b;
    a = na; b = nb;
}

// One block = 16 samples. Phase 1: wave 0 computes tanh(x@[W_pi1|W_v1]^T + b)
// for the 16-sample tile with two v_wmma_f32_16x16x4_f32 (K=8 split into 2x K=4),
// biases preloaded in the C accumulator. Result staged in LDS (16x16).
// Phase 2: 32 waves, one per (sample, head). 256-amplitude state lives in 8
// VGPRs/lane: reg bits = qubits 0-2, lane bits = qubits 3-7 (lane = b3*16+b4*8+b5*4+b6*2+b7).
__global__ __launch_bounds__(1024) void vqc_ac_kernel(
    const float* __restrict__ x,
    const float* __restrict__ W_pi1, const float* __restrict__ b_pi1,
    const float* __restrict__ qw_pi,
    const float* __restrict__ W_pi2, const float* __restrict__ b_pi2,
    const float* __restrict__ W_v1,  const float* __restrict__ b_v1,
    const float* __restrict__ qw_v,
    const float* __restrict__ W_v2,  const float* __restrict__ b_v2,
    float* __restrict__ out, int Bsz)
{
    __shared__ float hlds[16][16];   // [local sample][col: 0-7 pi latent, 8-15 v latent]
    const int lane = threadIdx.x & 31;
    const int wid  = threadIdx.x >> 5;
    const int base = blockIdx.x * 16;

    // ---------------- Phase 1: fused dual-head input GEMM via WMMA ----------------
    if (threadIdx.x < 32) {
        const int mrow = lane & 15;          // sample-in-tile / output column
        const int hi   = lane >> 4;          // lane half selects K pair {0,1} vs {2,3}
        const int k0   = hi * 2;
        int row = base + mrow; if (row >= Bsz) row = Bsz - 1;
        const float* xr = x + row * 8;
        // A (16x4 f32): VGPR0 = K0|K2, VGPR1 = K1|K3 across lane halves
        v2f A0; A0.x = xr[k0];     A0.y = xr[k0 + 1];
        v2f A1; A1.x = xr[k0 + 4]; A1.y = xr[k0 + 5];
        // B (4x16): col n<8 -> W_pi1[n][k], n>=8 -> W_v1[n-8][k]
        const float* Wrow = (mrow < 8) ? (W_pi1 + mrow * 8) : (W_v1 + (mrow - 8) * 8);
        v2f B0; B0.x = Wrow[k0];     B0.y = Wrow[k0 + 1];
        v2f B1; B1.x = Wrow[k0 + 4]; B1.y = Wrow[k0 + 5];
        // C preloaded with bias (same col for every row-register)
        float cb = (mrow < 8) ? b_pi1[mrow] : b_v1[mrow - 8];
        v8f acc;
#pragma unroll
        for (int i = 0; i < 8; ++i) acc[i] = cb;
        acc = __builtin_amdgcn_wmma_f32_16x16x4_f32(false, A0, false, B0, (short)0, acc, false, false);
        acc = __builtin_amdgcn_wmma_f32_16x16x4_f32(false, A1, false, B1, (short)0, acc, false, false);
        // D layout: reg r, lanes<16 -> sample r, lanes>=16 -> sample r+8; col = lane&15
#pragma unroll
        for (int r = 0; r < 8; ++r) hlds[r + hi * 8][mrow] = tanhf(acc[r]);
    }
    __syncthreads();

    // ---------------- Phase 2: one wave per (sample, head) VQC ----------------
    const int sid    = wid >> 1;
    const int head   = wid & 1;
    const int sample = base + sid;

    // Embedding RY on uniform H-state -> product state: amp = prod_q v_q[bit_q]
    float c_e[8], s_e[8];
#pragma unroll
    for (int q = 0; q < 8; ++q) {
        float h = 0.5f * hlds[sid][head * 8 + q];
        c_e[q] = cosf(h); s_e[q] = sinf(h);
    }
    const float r2 = 0.70710678118654752440f;
    float v0[8], v1[8];
#pragma unroll
    for (int q = 0; q < 8; ++q) {
        v0[q] = (c_e[q] - s_e[q]) * r2;   // target bit 0
        v1[q] = (c_e[q] + s_e[q]) * r2;   // target bit 1
    }
    float fl = (((lane >> 4) & 1) ? v1[3] : v0[3])
             * (((lane >> 3) & 1) ? v1[4] : v0[4])
             * (((lane >> 2) & 1) ? v1[5] : v0[5])
             * (((lane >> 1) & 1) ? v1[6] : v0[6])
             * (((lane     ) & 1) ? v1[7] : v0[7]);
    float s[8];
#pragma unroll
    for (int r = 0; r < 8; ++r) {
        float fr = (((r >> 2) & 1) ? v1[0] : v0[0])
                 * (((r >> 1) & 1) ? v1[1] : v0[1])
                 * (((r     ) & 1) ? v1[2] : v0[2]);
        s[r] = fl * fr;
    }

    const float* qw = head ? qw_v : qw_pi;
#pragma unroll
    for (int k = 0; k < 2; ++k) {
        // CNOT(0,1): reg-bit ctrl b0, reg-bit tgt b1 -> swap 4<->6, 5<->7
        { float t0 = s[4]; s[4] = s[6]; s[6] = t0; t0 = s[5]; s[5] = s[7]; s[7] = t0; }
        // CNOT(2,3): reg-bit ctrl b2 (odd r), lane-bit tgt b3 (mask 16)
#pragma unroll
        for (int r = 1; r < 8; r += 2) s[r] = __shfl_xor(s[r], 16, 32);
        // CNOT(4,5): lane ctrl mask 8, lane tgt mask 4
#pragma unroll
        for (int r = 0; r < 8; ++r) { float p = __shfl_xor(s[r], 4, 32); s[r] = (lane & 8) ? p : s[r]; }
        // CNOT(6,7): lane ctrl mask 2, lane tgt mask 1
#pragma unroll
        for (int r = 0; r < 8; ++r) { float p = __shfl_xor(s[r], 1, 32); s[r] = (lane & 2) ? p : s[r]; }
        // CNOT(1,2): reg ctrl b1, reg tgt b2 -> swap 2<->3, 6<->7
        { float t0 = s[2]; s[2] = s[3]; s[3] = t0; t0 = s[6]; s[6] = s[7]; s[7] = t0; }
        // CNOT(3,4): lane ctrl mask 16, lane tgt mask 8
#pragma unroll
        for (int r = 0; r < 8; ++r) { float p = __shfl_xor(s[r], 8, 32); s[r] = (lane & 16) ? p : s[r]; }
        // CNOT(5,6): lane ctrl mask 4, lane tgt mask 2
#pragma unroll
        for (int r = 0; r < 8; ++r) { float p = __shfl_xor(s[r], 2, 32); s[r] = (lane & 4) ? p : s[r]; }

        // Shared-weight RY layer
        float cw[8], sw[8];
#pragma unroll
        for (int q = 0; q < 8; ++q) { float h = 0.5f * qw[k * 8 + q]; cw[q] = cosf(h); sw[q] = sinf(h); }
        // q0 (reg bit2): pairs (r, r+4)
#pragma unroll
        for (int r = 0; r < 4; ++r) ry_pair(s[r], s[r + 4], cw[0], sw[0]);
        // q1 (reg bit1): pairs (r, r+2)
        ry_pair(s[0], s[2], cw[1], sw[1]); ry_pair(s[1], s[3], cw[1], sw[1]);
        ry_pair(s[4], s[6], cw[1], sw[1]); ry_pair(s[5], s[7], cw[1], sw[1]);
        // q2 (reg bit0): pairs (r, r+1)
        ry_pair(s[0], s[1], cw[2], sw[2]); ry_pair(s[2], s[3], cw[2], sw[2]);
        ry_pair(s[4], s[5], cw[2], sw[2]); ry_pair(s[6], s[7], cw[2], sw[2]);
        // q3..q7: lane-bit rotations, masks 16,8,4,2,1
#pragma unroll
        for (int q = 3; q < 8; ++q) {
            const int lm = 1 << (7 - q);
            float sgn = (lane & lm) ? sw[q] : -sw[q];
#pragma unroll
            for (int r = 0; r < 8; ++r) {
                float p = __shfl_xor(s[r], lm, 32);
                s[r] = cw[q] * s[r] + sgn * p;
            }
        }
    }

    // ---------------- Expectations <Z_q> ----------------
    float pr[8];
#pragma unroll
    for (int r = 0; r < 8; ++r) pr[r] = s[r] * s[r];
    float t  = ((pr[0]+pr[1])+(pr[2]+pr[3])) + ((pr[4]+pr[5])+(pr[6]+pr[7]));
    float u0 = ((pr[0]+pr[1])+(pr[2]+pr[3])) - ((pr[4]+pr[5])+(pr[6]+pr[7]));   // q0: reg bit2
    float u1 = ((pr[0]+pr[1])+(pr[4]+pr[5])) - ((pr[2]+pr[3])+(pr[6]+pr[7]));   // q1: reg bit1
    float u2 = ((pr[0]+pr[2])+(pr[4]+pr[6])) - ((pr[1]+pr[3])+(pr[5]+pr[7]));   // q2: reg bit0
    // Plain all-reduce for u*, Walsh-Hadamard butterfly for t (lane 2^b holds signed sum for lane-bit b)
#pragma unroll
    for (int mm = 1; mm < 32; mm <<= 1) {
        u0 += __shfl_xor(u0, mm, 32);
        u1 += __shfl_xor(u1, mm, 32);
        u2 += __shfl_xor(u2, mm, 32);
        float tq = __shfl_xor(t, mm, 32);
        t = (lane & mm) ? (tq - t) : (t + tq);
    }
    // Gather: lane q holds tanh(<Z_q>). q3..q7 live at WHT lanes 16,8,4,2,1.
    int src = (lane >= 3 && lane < 8) ? (1 << (7 - lane)) : 0;
    float tg = __shfl(t, src, 32);
    float e = (lane == 0) ? u0 : (lane == 1) ? u1 : (lane == 2) ? u2 : tg;
    float g = tanhf(e);

    // ---------------- Output heads ----------------
    if (head == 0) {
#pragma unroll
        for (int a = 0; a < 4; ++a) {
            float pt = (lane < 8) ? g * W_pi2[a * 8 + (lane & 7)] : 0.0f;
            pt += __shfl_xor(pt, 1, 32);
            pt += __shfl_xor(pt, 2, 32);
            pt += __shfl_xor(pt, 4, 32);
            if (lane == 0 && sample < Bsz) out[sample * 4 + a] = pt + b_pi2[a];
        }
    } else {
        float pt = (lane < 8) ? g * W_v2[lane & 7] : 0.0f;
        pt += __shfl_xor(pt, 1, 32);
        pt += __shfl_xor(pt, 2, 32);
        pt += __shfl_xor(pt, 4, 32);
        if (lane == 0 && sample < Bsz) out[(size_t)Bsz * 4 + sample] = pt + b_v2[0];
    }
}

extern "C" void kernel_launch(void* const* d_in, const int* in_sizes, int n_in,
                              void* d_out, int out_size, void* d_ws, size_t ws_size,
                              hipStream_t stream) {
    const float* x     = (const float*)d_in[0];
    const float* W_pi1 = (const float*)d_in[1];
    const float* b_pi1 = (const float*)d_in[2];
    const float* qw_pi = (const float*)d_in[3];
    const float* W_pi2 = (const float*)d_in[4];
    const float* b_pi2 = (const float*)d_in[5];
    const float* W_v1  = (const float*)d_in[6];
    const float* b_v1  = (const float*)d_in[7];
    const float* qw_v  = (const float*)d_in[8];
    const float* W_v2  = (const float*)d_in[9];
    const float* b_v2  = (const float*)d_in[10];
    float* out = (float*)d_out;
    int Bsz = in_sizes[0] / 8;
    int blocks = (Bsz + 15) / 16;
    vqc_ac_kernel<<<blocks, 1024, 0, stream>>>(x, W_pi1, b_pi1, qw_pi, W_pi2, b_pi2,
                                               W_v1, b_v1, qw_v, W_v2, b_v2, out, Bsz);
}